// CRPS_8916352106762
// MI455X (gfx1250) — compile-verified
//
#include <hip/hip_runtime.h>

// ---------------------------------------------------------------------------
// CRPS (log-normal survival) loss + mean over B=8.4M rows.
// Compute-bound streaming kernel: VALU-minimized f32 math.
//   - I(Y) and I_(-Y) share one erf/exp evaluation set (z vs -z symmetry)
//   - erf via A&S 7.1.26 reusing exp(-z^2/2) already needed for the pdf:
//       1 v_exp_f32 + 2 v_rcp_f32 + ~12 FMA per z-evaluation
// WMMA (v_wmma_f32_16x16x4_f32) performs the cross-lane wave reduction.
// ---------------------------------------------------------------------------

typedef __attribute__((ext_vector_type(2))) float v2f;
typedef __attribute__((ext_vector_type(8))) float v8f;

#define INV_SQRT_PI   0.5641895835477563f
#define INV_SQRT_2PI  0.3989422804014327f
#define SQRT1_2       0.7071067811865476f
#define AGE_MAX       120.0f

// ------------------------- wave / block reduction --------------------------

// Full 32-lane wave sum via the matrix pipe:
// A (16x4 f32): a0=acc, a1=0 -> A[m][0]=acc_m, A[m][2]=acc_{m+16}
// B = ones     -> D[m][n] = acc_m + acc_{m+16}
// Per-lane sum of the 8 D VGPRs covers 8 distinct rows; the xor-16 partner
// lane covers the complementary 8 -> one SWAPX16 swizzle-add finishes it.
__device__ __forceinline__ float wave_sum_wmma(float acc) {
    v2f a; a[0] = acc;  a[1] = 0.0f;
    v2f b; b[0] = 1.0f; b[1] = 1.0f;
    v8f c = {};
    v8f d = __builtin_amdgcn_wmma_f32_16x16x4_f32(
        false, a, false, b, (short)0, c, false, false);
    float s = ((d[0] + d[1]) + (d[2] + d[3])) + ((d[4] + d[5]) + (d[6] + d[7]));
    // ds_swizzle SWAPX16: offset = xor(0x10)<<10 | and(0x1f) = 0x401F
    s += __int_as_float(__builtin_amdgcn_ds_swizzle(__float_as_int(s), 0x401F));
    return s;
}

// Deterministic block sum (256 threads = 8 waves). Result valid in thread 0.
__device__ __forceinline__ float block_sum_256(float acc, float* smem) {
    const float w = wave_sum_wmma(acc);
    if ((threadIdx.x & 31) == 0) smem[threadIdx.x >> 5] = w;
    __syncthreads();
    float b = 0.0f;
    if (threadIdx.x == 0) {
#pragma unroll
        for (int i = 0; i < 8; ++i) b += smem[i];
    }
    return b;
}

// ------------------------------ CRPS math ----------------------------------

// erf(x) for x >= 0 given ex2 = exp(-x*x). A&S 7.1.26, |abs err| < 1.5e-7.
__device__ __forceinline__ float erf_pos(float x, float ex2) {
    const float t = __builtin_amdgcn_rcpf(__fmaf_rn(0.3275911f, x, 1.0f));
    float p = __fmaf_rn(1.061405429f, t, -1.453152027f);
    p = __fmaf_rn(p, t, 1.421413741f);
    p = __fmaf_rn(p, t, -0.284496736f);
    p = __fmaf_rn(p, t, 0.254829592f);
    return __fmaf_rn(-p * t, ex2, 1.0f);
}

struct ZStats { float c, c2, p0; };  // ndtr(z), ndtr(z*sqrt2), pdf(z)

__device__ __forceinline__ ZStats zeval(float z) {
    const float az = fabsf(z);
    const float e1 = __expf(-0.5f * z * z);     // exp(-z^2/2)  (v_exp_f32)
    const float e2 = e1 * e1;                   // exp(-z^2) for free
    const float Eh = erf_pos(az * SQRT1_2, e1); // erf(|z|/sqrt2)
    const float Ef = erf_pos(az, e2);           // erf(|z|)
    ZStats r;
    r.c  = __fmaf_rn(0.5f, copysignf(Eh, z), 0.5f);
    r.c2 = __fmaf_rn(0.5f, copysignf(Ef, z), 0.5f);
    r.p0 = INV_SQRT_2PI * e1;
    return r;
}

// I_norm(mu, s^2, y) evaluated from z-stats: s*(z*c^2 + 2*c*p0 - K*c2)
__device__ __forceinline__ float I_from(float z, const ZStats& st, float s) {
    const float term = __fmaf_rn(z * st.c, st.c,
                       __fmaf_rn(2.0f * st.c, st.p0, -INV_SQRT_PI * st.c2));
    return s * term;
}
// I_norm(-mu, s^2, -y) from the SAME stats (z -> -z symmetry).
__device__ __forceinline__ float I_from_neg(float z, const ZStats& st, float s) {
    const float c  = 1.0f - st.c;
    const float c2 = 1.0f - st.c2;
    const float term = __fmaf_rn(-z * c, c,
                       __fmaf_rn(2.0f * c, st.p0, -INV_SQRT_PI * c2));
    return s * term;
}

// ------------------------------- kernels -----------------------------------

__global__ void __launch_bounds__(256)
crps_partial(const float2* __restrict__ pred, const float2* __restrict__ tgt,
             const float* __restrict__ ages, const int* __restrict__ use_intvl_p,
             float* __restrict__ part, int n) {
    const int use_intvl = *use_intvl_p;           // uniform scalar load
    const int stride = gridDim.x * blockDim.x;
    const int i0 = blockIdx.x * blockDim.x + threadIdx.x;
    float acc = 0.0f;

    if (use_intvl) {
        for (int i = i0; i < n; i += stride) {
            const float2 pp = pred[i];
            const float mu    = pp.x;
            const float s     = __expf( 0.5f * pp.y);   // sqrt(exp(p1))
            const float inv_s = __expf(-0.5f * pp.y);   // exact 1/s
            const float2 tg = tgt[i];
            const float cens = tg.y;

            const float zY = (__logf(tg.x) - mu) * inv_s;
            const float zT = (mu - __logf(AGE_MAX - ages[i])) * inv_s;

            const ZStats sY = zeval(zY);
            const ZStats sT = zeval(zT);

            const float iY  = I_from(zY, sY, s);        // I(Y)
            const float iNY = I_from_neg(zY, sY, s);    // I_(-Y), shared stats
            const float iNT = I_from(zT, sT, s);        // I_(-T)

            acc += iY + iNT + (1.0f - cens) * (iNY - iNT);
        }
    } else {
        for (int i = i0; i < n; i += stride) {
            const float2 pp = pred[i];
            const float mu    = pp.x;
            const float s     = __expf( 0.5f * pp.y);
            const float inv_s = __expf(-0.5f * pp.y);
            const float2 tg = tgt[i];
            const float cens = tg.y;

            const float zY = (__logf(tg.x) - mu) * inv_s;
            const ZStats sY = zeval(zY);

            const float iY  = I_from(zY, sY, s);
            const float iNY = I_from_neg(zY, sY, s);

            acc += iY + (1.0f - cens) * iNY;
        }
    }

    __shared__ float smem[8];
    const float b = block_sum_256(acc, smem);
    if (threadIdx.x == 0) part[blockIdx.x] = b;
}

__global__ void __launch_bounds__(256)
crps_final(const float* __restrict__ part, int npart,
           float* __restrict__ out, float scale) {
    float acc = 0.0f;
    for (int i = threadIdx.x; i < npart; i += 256) acc += part[i];
    __shared__ float smem[8];
    const float b = block_sum_256(acc, smem);
    if (threadIdx.x == 0) out[0] = b * scale;     // plain store: deterministic
}

// ------------------------------- launcher ----------------------------------

extern "C" void kernel_launch(void* const* d_in, const int* in_sizes, int n_in,
                              void* d_out, int out_size, void* d_ws, size_t ws_size,
                              hipStream_t stream) {
    const float2* pred      = (const float2*)d_in[0];   // (B,2) f32
    const float2* tgt       = (const float2*)d_in[1];   // (B,2) f32
    const float*  ages      = (const float*)d_in[2];    // (B,)  f32
    const int*    use_intvl = (const int*)d_in[3];      // scalar
    float* out  = (float*)d_out;
    float* part = (float*)d_ws;

    const int n = in_sizes[2];                          // B

    int nblocks = 2048;                                 // ~8 blocks/WGP-class
    const size_t max_part = ws_size / sizeof(float);
    if ((size_t)nblocks > max_part) nblocks = (int)max_part;
    if (nblocks < 1) nblocks = 1;

    crps_partial<<<nblocks, 256, 0, stream>>>(pred, tgt, ages, use_intvl, part, n);
    crps_final<<<1, 256, 0, stream>>>(part, nblocks, out, 1.0f / (float)n);
}